// StyleModLayer_46935402611224
// MI455X (gfx1250) — compile-verified
//
#include <hip/hip_runtime.h>
#include <hip/hip_bf16.h>

typedef __attribute__((ext_vector_type(2))) float v2f;
typedef __attribute__((ext_vector_type(8))) float v8f;

#define NB        16
#define IN_C      64
#define OUT_C     64
#define HW        16384   // 128*128
#define STYLE_DIM 512
#define LDS_STRIDE 68     // pad 64 -> 68: (4*o + k) % 64 distinct banks, rows 8B-aligned

// ---------------------------------------------------------------------------
// Kernel 1: s = style@geo_w^T + geo_b ; mod = s+1 ; w = weight*mod ;
//           demod = rsqrt(sum_i w^2 + eps) ; wmod = w*demod   -> d_ws
// Tiny: 16 blocks x 64 threads. Thread t computes mod[t], then row o=t.
// ---------------------------------------------------------------------------
__global__ __launch_bounds__(64) void style_modweight_kernel(
    const float* __restrict__ style,   // (B, 512)
    const float* __restrict__ weight,  // (64, 64) flattened from (1,64,64,1,1)
    const float* __restrict__ geo_w,   // (64, 512)
    const float* __restrict__ geo_b,   // (64,)
    float* __restrict__ wmod)          // (B, 64, 64)
{
    __shared__ float mod[IN_C];
    const int b = blockIdx.x;
    const int t = threadIdx.x;              // 0..63

    // mod[t] = dot(style[b,:], geo_w[t,:]) + geo_b[t] + 1
    const float* st = style + b * STYLE_DIM;
    const float* gw = geo_w + t * STYLE_DIM;
    float acc = 0.0f;
#pragma unroll 8
    for (int j = 0; j < STYLE_DIM; ++j)
        acc = fmaf(st[j], gw[j], acc);
    mod[t] = acc + geo_b[t] + 1.0f;
    __syncthreads();

    // row o = t: modulate, demodulate, write
    const int o = t;
    const float* wrow = weight + o * IN_C;
    float ss = 0.0f;
#pragma unroll
    for (int i = 0; i < IN_C; ++i) {
        float ww = wrow[i] * mod[i];
        ss = fmaf(ww, ww, ss);
    }
    const float demod = rsqrtf(ss + 1e-8f);
    float* dst = wmod + ((size_t)b * OUT_C + o) * IN_C;
#pragma unroll
    for (int i = 0; i < IN_C; ++i)
        dst[i] = wrow[i] * mod[i] * demod;
}

// ---------------------------------------------------------------------------
// Kernel 2: out[b,o,p] = sum_i wmod[b,o,i] * x[b,i,p] + bias[o]
// Per sample: (64x64) x (64x16384) GEMM via V_WMMA_F32_16X16X4_F32.
// Block = 256 thr (8 waves); wave -> 16 pixels (N), all 4 M-tiles, K=64.
//
// WMMA f32 16x16x4 VGPR layouts (ISA 7.12.2):
//   A (16x4, MxK): lane L -> M=L&15 ; VGPR0 = K {0 | 2 for hi half}, VGPR1 = K {1 | 3}
//   B (4x16, KxN): lane L -> N=L&15 ; VGPR0 = K {0 | 2}, VGPR1 = K {1 | 3}
//   D (16x16):     VGPR v: lanes 0-15 -> M=v, lanes 16-31 -> M=v+8 ; N=lane&15
// ---------------------------------------------------------------------------
__global__ __launch_bounds__(256) void styled_conv1x1_wmma_kernel(
    const float* __restrict__ x,      // (B, 64, HW)
    const float* __restrict__ wmod,   // (B, 64, 64)
    const float* __restrict__ bias,   // (64,)
    float* __restrict__ out)          // (B, 64, HW)
{
    __shared__ float wl[OUT_C * LDS_STRIDE];
    __shared__ float bl[OUT_C];

    const int b   = blockIdx.y;
    const int tid = threadIdx.x;

    // Stage per-sample 64x64 weight matrix (+bias) into LDS, padded rows.
    const float* wsrc = wmod + (size_t)b * OUT_C * IN_C;
    for (int idx = tid; idx < OUT_C * IN_C; idx += 256) {
        const int o = idx >> 6, i = idx & 63;
        wl[o * LDS_STRIDE + i] = wsrc[idx];
    }
    if (tid < OUT_C) bl[tid] = bias[tid];
    __syncthreads();

    const int wave = tid >> 5;
    const int lane = tid & 31;
    const int n    = lane & 15;      // column within 16-pixel tile
    const int hi   = lane >> 4;      // half-wave selector (K group)

    const int pix = blockIdx.x * 128 + wave * 16 + n;
    const float* xb = x + (size_t)b * IN_C * HW + pix;

    // Load the full K=64 strip of B (x) into 16 register pairs, read-once.
    v2f Breg[16];
#pragma unroll
    for (int s = 0; s < 16; ++s) {
        const int k0 = 4 * s + 2 * hi;
        Breg[s].x = xb[(size_t)k0 * HW];
        Breg[s].y = xb[(size_t)(k0 + 1) * HW];
    }

    float* ob = out + (size_t)b * OUT_C * HW + pix;

#pragma unroll
    for (int m = 0; m < 4; ++m) {
        v8f acc = {};
        // A row for this lane: o = m*16 + n, K base offset 2*hi (8B-aligned -> ds_load_b64)
        const float* arow = &wl[(m * 16 + n) * LDS_STRIDE + 2 * hi];
#pragma unroll
        for (int s = 0; s < 16; ++s) {
            v2f a;
            a.x = arow[4 * s];
            a.y = arow[4 * s + 1];
            // D = A x B + C  (fp32 matrix pipe)
            acc = __builtin_amdgcn_wmma_f32_16x16x4_f32(
                      /*neg_a=*/false, a, /*neg_b=*/false, Breg[s],
                      /*c_mod=*/(short)0, acc,
                      /*reuse_a=*/false, /*reuse_b=*/false);
        }
        // Store D tile + bias. VGPR v holds rows m*16+v (lanes 0-15) / m*16+v+8 (16-31).
#pragma unroll
        for (int v = 0; v < 8; ++v) {
            const int o = m * 16 + v + 8 * hi;
            ob[(size_t)o * HW] = acc[v] + bl[o];
        }
    }
}

// ---------------------------------------------------------------------------
extern "C" void kernel_launch(void* const* d_in, const int* in_sizes, int n_in,
                              void* d_out, int out_size, void* d_ws, size_t ws_size,
                              hipStream_t stream) {
    const float* x      = (const float*)d_in[0];  // (16,64,128,128)
    const float* style  = (const float*)d_in[1];  // (16,512)
    const float* weight = (const float*)d_in[2];  // (1,64,64,1,1)
    const float* bias   = (const float*)d_in[3];  // (1,64)
    const float* geo_w  = (const float*)d_in[4];  // (64,512)
    const float* geo_b  = (const float*)d_in[5];  // (64,)
    float* out  = (float*)d_out;
    float* wmod = (float*)d_ws;                   // (16,64,64) = 256 KB scratch

    style_modweight_kernel<<<NB, 64, 0, stream>>>(style, weight, geo_w, geo_b, wmod);

    dim3 grid(HW / 128, NB);  // 128 pixels per block (8 waves x 16)
    styled_conv1x1_wmma_kernel<<<grid, 256, 0, stream>>>(x, wmod, bias, out);
}